// Loudness_86681029967942
// MI455X (gfx1250) — compile-verified
//
#include <hip/hip_runtime.h>
#include <math.h>

typedef __attribute__((ext_vector_type(2))) float v2f;
typedef __attribute__((ext_vector_type(8))) float v8f;

#define TSAMP 240000
#define NBATCH 64
#define SEGLEN 960          // samples per segment
#define NSEG   250          // 250*960 = 240000
#define NBLK   (SEGLEN/16)  // 60 WMMA tiles per segment

// ---------------------------------------------------------------------------
// Phase 1: per-segment zero-state biquad via blocked WMMA.
// grid = (NSEG, 4 batch-groups), block = 32 (one wave32, EXEC all ones).
// Writes unclipped zero-state outputs y0 and final (z1,z2) per (seg,batch).
// NOTE: xin/y0 may alias (in-place second biquad) -> no __restrict__ there.
// ---------------------------------------------------------------------------
__global__ __launch_bounds__(32)
void k_seg_scan(const float* xin, float* y0,
                float* __restrict__ zfin,
                const float* __restrict__ bco, const float* __restrict__ aco,
                int row)
{
    const int tid = threadIdx.x;
    const int n   = tid & 15;     // batch column within group / matrix N,M index
    const int hi  = tid >> 4;     // lane half
    const int seg = blockIdx.x;
    const int grp = blockIdx.y;

    const float b0 = bco[3*row+0], b1 = bco[3*row+1], b2 = bco[3*row+2];
    const float a1 = aco[3*row+1], a2 = aco[3*row+2];

    // Impulse response h[0..15] and homogeneous state responses p/q[0..15]
    __shared__ float hs[16], ps[16], qs[16];
    if (tid == 0) {
        float z1 = 0.f, z2 = 0.f, xi = 1.f;
        for (int t = 0; t < 16; ++t) {
            float y = b0*xi + z1;
            hs[t] = y;
            float z1n = b1*xi + z2 - a1*y;
            z2 = b2*xi - a2*y;
            z1 = z1n;
            xi = 0.f;
        }
        ps[0] = 1.f;  qs[0] = 0.f;
        ps[1] = -a1;  qs[1] = 1.f;
        float pm2 = 1.f, pm1 = -a1, qm2 = 0.f, qm1 = 1.f;
        for (int t = 2; t < 16; ++t) {
            float p = -a1*pm1 - a2*pm2;
            float q = -a1*qm1 - a2*qm2;
            ps[t] = p; qs[t] = q;
            pm2 = pm1; pm1 = p; qm2 = qm1; qm1 = q;
        }
    }
    __syncthreads();

    // A-matrix (Toeplitz T, column slices of K=4).  Lane layout (16x4 f32):
    // lanes 0-15: M=lane,K={0,1}+4s ; lanes 16-31: M=lane-16,K={2,3}+4s
    v2f A[4];
#pragma unroll
    for (int s = 0; s < 4; ++s) {
        int k0 = 4*s + 2*hi;
        A[s].x = (n >= k0    ) ? hs[n - k0    ] : 0.f;
        A[s].y = (n >= k0 + 1) ? hs[n - k0 - 1] : 0.f;
    }
    // Per-lane state-response rows for the C/D layout (row m = r + 8*hi)
    float pv[8], qv[8];
#pragma unroll
    for (int r = 0; r < 8; ++r) { pv[r] = ps[r + 8*hi]; qv[r] = qs[r + 8*hi]; }

    const size_t rowbase = (size_t)(grp*16 + n) * TSAMP + (size_t)seg * SEGLEN;
    const float* xcol = xin + rowbase;
    float*       ycol = y0  + rowbase;

    float z1v = 0.f, z2v = 0.f;   // lane holds state of batch column n

    for (int blk = 0; blk < NBLK; ++blk) {
        const int t0 = blk * 16;
        // B-matrix slices (4x16 f32, mirrors A layout): lane holds
        // X[k0][col n], X[k0+1][col n] with k0 = 4s + 2*hi.
        v2f B[4];
#pragma unroll
        for (int s = 0; s < 4; ++s) {
            int k0 = 4*s + 2*hi;
            B[s] = *(const v2f*)(xcol + t0 + k0);
        }
        // C init = incoming-state correction, then 4 WMMAs accumulate T.X
        v8f C;
#pragma unroll
        for (int r = 0; r < 8; ++r) C[r] = pv[r]*z1v + qv[r]*z2v;
#pragma unroll
        for (int s = 0; s < 4; ++s)
            C = __builtin_amdgcn_wmma_f32_16x16x4_f32(
                    false, A[s], false, B[s], (short)0, C, false, false);

        // Recover DF2T state from last two inputs/outputs (per batch column)
        float x15 = __shfl(B[3].y, 16 + n, 32);  // k=15 lives in hi half of B[3].y
        float x14 = __shfl(B[3].x, 16 + n, 32);  // k=14 lives in hi half of B[3].x
        float y15 = __shfl(C[7],   16 + n, 32);  // row 15 = VGPR7, hi half
        float y14 = __shfl(C[6],   16 + n, 32);  // row 14 = VGPR6, hi half
        float z1n = b1*x15 + b2*x14 - a2*y14 - a1*y15;
        z2v = b2*x15 - a2*y15;
        z1v = z1n;

        // Store unclipped zero-state outputs (rows r+8*hi of column n)
#pragma unroll
        for (int r = 0; r < 8; ++r) ycol[t0 + r + 8*hi] = C[r];
    }

    if (hi == 0) {
        size_t zi = ((size_t)seg * NBATCH + grp*16 + n) * 2;
        zfin[zi + 0] = z1v;
        zfin[zi + 1] = z2v;
    }
}

// ---------------------------------------------------------------------------
// Phase 2: serial segment-state combine (64 threads, one per batch row) and
// build p/q correction table for offsets [0, SEGLEN).
// ---------------------------------------------------------------------------
__global__ __launch_bounds__(64)
void k_combine(const float* __restrict__ bco, const float* __restrict__ aco,
               int row, const float* __restrict__ zfin,
               float* __restrict__ zin, float* __restrict__ pq)
{
    const int tid = threadIdx.x;
    const float a1 = aco[3*row+1], a2 = aco[3*row+2];
    (void)bco;

    if (tid == 0) {  // p/q homogeneous-response table, interleaved [p,q]
        pq[0] = 1.f;  pq[1] = 0.f;
        pq[2] = -a1;  pq[3] = 1.f;
        float pm2 = 1.f, pm1 = -a1, qm2 = 0.f, qm1 = 1.f;
        for (int t = 2; t < SEGLEN; ++t) {
            float p = -a1*pm1 - a2*pm2;
            float q = -a1*qm1 - a2*qm2;
            pq[2*t] = p; pq[2*t+1] = q;
            pm2 = pm1; pm1 = p; qm2 = qm1; qm1 = q;
        }
    }

    // ML = M^SEGLEN, M = [[-a1, 1], [-a2, 0]] (redundant per thread, uniform)
    float m00 = 1.f, m01 = 0.f, m10 = 0.f, m11 = 1.f;
    for (int t = 0; t < SEGLEN; ++t) {
        float n00 = -a1*m00 + m10, n01 = -a1*m01 + m11;
        float n10 = -a2*m00,       n11 = -a2*m01;
        m00 = n00; m01 = n01; m10 = n10; m11 = n11;
    }

    const int b = tid;
    float z1 = 0.f, z2 = 0.f;
    for (int s = 0; s < NSEG; ++s) {
        size_t zi = ((size_t)s * NBATCH + b) * 2;
        zin[zi + 0] = z1;
        zin[zi + 1] = z2;
        float f1 = zfin[zi + 0], f2 = zfin[zi + 1];
        float z1n = m00*z1 + m01*z2 + f1;
        float z2n = m10*z1 + m11*z2 + f2;
        z1 = z1n; z2 = z2n;
    }
}

// ---------------------------------------------------------------------------
// Phase 3a: y = clip(y0 + p*z1 + q*z2)  (in place, coalesced over t)
// ---------------------------------------------------------------------------
__global__ __launch_bounds__(256)
void k_correct_clip(float* __restrict__ y, const float* __restrict__ zin,
                    const float* __restrict__ pq)
{
    const int b = blockIdx.y;
    const int t = blockIdx.x * 256 + threadIdx.x;
    if (t >= TSAMP) return;
    const int s   = t / SEGLEN;
    const int off = t - s * SEGLEN;
    const size_t zi = ((size_t)s * NBATCH + b) * 2;
    float v = y[(size_t)b * TSAMP + t]
            + pq[2*off] * zin[zi] + pq[2*off+1] * zin[zi+1];
    v = fminf(1.f, fmaxf(-1.f, v));
    y[(size_t)b * TSAMP + t] = v;
}

// ---------------------------------------------------------------------------
// Phase 3b (final biquad): correct + clip + square + block reduce + atomic
// ---------------------------------------------------------------------------
__global__ __launch_bounds__(256)
void k_correct_sq(const float* __restrict__ y, const float* __restrict__ zin,
                  const float* __restrict__ pq, float* __restrict__ acc)
{
    const int b = blockIdx.y;
    const int t = blockIdx.x * 256 + threadIdx.x;
    float v = 0.f;
    if (t < TSAMP) {
        const int s   = t / SEGLEN;
        const int off = t - s * SEGLEN;
        const size_t zi = ((size_t)s * NBATCH + b) * 2;
        v = y[(size_t)b * TSAMP + t]
          + pq[2*off] * zin[zi] + pq[2*off+1] * zin[zi+1];
        v = fminf(1.f, fmaxf(-1.f, v));
        v = v * v;
    }
#pragma unroll
    for (int o = 16; o > 0; o >>= 1) v += __shfl_down(v, o, 32);
    __shared__ float red[8];
    const int wid = threadIdx.x >> 5, lane = threadIdx.x & 31;
    if (lane == 0) red[wid] = v;
    __syncthreads();
    if (threadIdx.x == 0) {
        float sacc = 0.f;
#pragma unroll
        for (int i = 0; i < 8; ++i) sacc += red[i];
        atomicAdd(&acc[b], sacc);
    }
}

__global__ __launch_bounds__(64)
void k_init_acc(float* __restrict__ acc) { acc[threadIdx.x] = 0.f; }

__global__ __launch_bounds__(64)
void k_finalize(const float* __restrict__ acc, float* __restrict__ out)
{
    const int b = threadIdx.x;
    float ms = acc[b] / (float)TSAMP;
    out[b] = -0.691f + 10.f * log10f(ms + 1e-8f);
}

// ---------------------------------------------------------------------------
extern "C" void kernel_launch(void* const* d_in, const int* in_sizes, int n_in,
                              void* d_out, int out_size, void* d_ws, size_t ws_size,
                              hipStream_t stream)
{
    const float* x   = (const float*)d_in[0];
    const float* bco = (const float*)d_in[1];
    const float* aco = (const float*)d_in[2];
    float* out = (float*)d_out;

    // Workspace carve-up (floats)
    float* y0  = (float*)d_ws;                       // NBATCH*TSAMP
    float* zf  = y0 + (size_t)NBATCH * TSAMP;        // NSEG*NBATCH*2
    float* zi  = zf + (size_t)NSEG * NBATCH * 2;     // NSEG*NBATCH*2
    float* pq  = zi + (size_t)NSEG * NBATCH * 2;     // SEGLEN*2
    float* acc = pq + (size_t)SEGLEN * 2;            // NBATCH

    dim3 segGrid(NSEG, 4);
    dim3 elemGrid((TSAMP + 255) / 256, NBATCH);

    k_init_acc<<<1, 64, 0, stream>>>(acc);

    // Biquad 1 (high-shelf)
    k_seg_scan<<<segGrid, 32, 0, stream>>>(x, y0, zf, bco, aco, 0);
    k_combine<<<1, 64, 0, stream>>>(bco, aco, 0, zf, zi, pq);
    k_correct_clip<<<elemGrid, 256, 0, stream>>>(y0, zi, pq);

    // Biquad 2 (high-pass), in place over y0
    k_seg_scan<<<segGrid, 32, 0, stream>>>(y0, y0, zf, bco, aco, 1);
    k_combine<<<1, 64, 0, stream>>>(bco, aco, 1, zf, zi, pq);
    k_correct_sq<<<elemGrid, 256, 0, stream>>>(y0, zi, pq, acc);

    k_finalize<<<1, 64, 0, stream>>>(acc, out);
}